// TemporalGNN_23811298689805
// MI455X (gfx1250) — compile-verified
//
#include <hip/hip_runtime.h>
#include <hip/hip_bf16.h>
#include <math.h>

// ---------------------------------------------------------------------------
// Problem constants (match reference)
// ---------------------------------------------------------------------------
#define GN1 100000
#define GN2 20000
#define GC_IN 8
#define GT 12
#define GH 64            // H1 == H2 == 64
#define GSECOND 16

typedef float v2f __attribute__((ext_vector_type(2)));
typedef float v8f __attribute__((ext_vector_type(8)));

enum { ACT_NONE = 0, ACT_SIGMOID = 1, ACT_TANH = 2, ACT_RELU = 3 };

// ---------------------------------------------------------------------------
// Generic fp32 WMMA GEMM:
//   C[N x OC] = act( concat(A1[N x K1], A2[N x K2]) @ W[(K1+K2) x OC] + bias )
// A1/A2 have arbitrary (row,col) strides (slices of (N,C,T) tensors, virtual
// concats). One block = 32 rows x 64 cols; 4 waves each own a 16-col slice
// and two 16-row tiles (two accumulators -> two back-to-back WMMAs per B
// fragment). Both A (32 x K) and W (K x 64, zero-padded, stored col-major)
// are staged in LDS once, so the K-loop is pure ds_load_b64 + v_wmma with no
// global traffic and no divergence (EXEC all-1s as WMMA requires).
// N must be a multiple of 32 (100000 and 20000 both are).
// ---------------------------------------------------------------------------
__global__ __launch_bounds__(128) void wmma_gemm_kernel(
    const float* __restrict__ A1, int r1s, int c1s, int K1,
    const float* __restrict__ A2, int r2s, int c2s, int K2,
    const float* __restrict__ W, const float* __restrict__ bias,
    float* __restrict__ Cout, int OC, int act) {
  __shared__ float Alds[32 * 132];  // K <= 128 -> KSTR <= 132
  __shared__ float Wlds[64 * 132];  // col-major: Wlds[col*KSTR + k]

  const int K = K1 + K2;
  const int KP = (K + 3) & ~3;  // K padded to multiple of 4
  const int KSTR = KP + 4;      // row stride: mult of 4 (b64-aligned), not 64
  const int lane = threadIdx.x; // 0..31
  const int wv = threadIdx.y;   // 0..3: column tile
  const int tid = wv * 32 + lane;
  const long row0 = (long)blockIdx.x * 32;

  // Stage the 32 x K A-tile (concat A1|A2, zero pad to KP).
  for (int idx = tid; idx < 32 * KP; idx += 128) {
    int r = idx / KP;
    int kk = idx - r * KP;
    float v = 0.0f;
    if (kk < K1)
      v = A1[(row0 + r) * (long)r1s + (long)kk * c1s];
    else if (kk < K)
      v = A2[(row0 + r) * (long)r2s + (long)(kk - K1) * c2s];
    Alds[r * KSTR + kk] = v;
  }
  // Stage W (K x OC) zero-padded to KP x 64, transposed to col-major so the
  // B fragment is a contiguous 8-byte LDS read.
  for (int idx = tid; idx < 64 * KP; idx += 128) {
    int kk = idx >> 6;       // coalesced global read along the 64-col rows
    int col = idx & 63;
    float v = (kk < K && col < OC) ? W[(long)kk * OC + col] : 0.0f;
    Wlds[col * KSTR + kk] = v;
  }
  __syncthreads();

  const int col0 = wv * 16;
  if (col0 >= ((OC + 15) & ~15)) return;  // wave-uniform exit (OC < 64)

  const bool hi = lane >= 16;
  const int l = lane & 15;
  const int bcol = col0 + l;
  const int koff = hi ? 2 : 0;
  const float* Arow0 = &Alds[l * KSTR];         // rows 0..15 of the tile
  const float* Arow1 = &Alds[(l + 16) * KSTR];  // rows 16..31
  const float* Bcol = &Wlds[bcol * KSTR];

  v8f acc0 = {0.f, 0.f, 0.f, 0.f, 0.f, 0.f, 0.f, 0.f};
  v8f acc1 = {0.f, 0.f, 0.f, 0.f, 0.f, 0.f, 0.f, 0.f};
#pragma unroll 2
  for (int k = 0; k < KP; k += 4) {
    const int ka = k + koff;
    // A fragments: 16x4 f32 (lanes 0-15: K=k..k+1, lanes 16-31: K=k+2..k+3)
    v2f a0 = *(const v2f*)&Arow0[ka];
    v2f a1 = *(const v2f*)&Arow1[ka];
    // B fragment: 4x16 f32, mirrored layout, contiguous in LDS
    v2f b = *(const v2f*)&Bcol[ka];
    acc0 = __builtin_amdgcn_wmma_f32_16x16x4_f32(false, a0, false, b, (short)0,
                                                 acc0, false, false);
    acc1 = __builtin_amdgcn_wmma_f32_16x16x4_f32(false, a1, false, b, (short)0,
                                                 acc1, false, false);
  }

  if (bcol >= OC) return;  // per-lane store guard (only OC=12 case)
  const float bv = bias ? bias[bcol] : 0.0f;
  const int mb = hi ? 8 : 0;
#pragma unroll
  for (int i = 0; i < 8; ++i) {
    float v0 = acc0[i] + bv;
    float v1 = acc1[i] + bv;
    if (act == ACT_SIGMOID) {
      v0 = 1.0f / (1.0f + __expf(-v0));
      v1 = 1.0f / (1.0f + __expf(-v1));
    } else if (act == ACT_TANH) {
      v0 = tanhf(v0);
      v1 = tanhf(v1);
    } else if (act == ACT_RELU) {
      v0 = fmaxf(v0, 0.0f);
      v1 = fmaxf(v1, 0.0f);
    }
    Cout[(row0 + mb + i) * (long)OC + bcol] = v0;
    Cout[(row0 + 16 + mb + i) * (long)OC + bcol] = v1;
  }
}

// ---------------------------------------------------------------------------
// Elementwise / scatter kernels
// ---------------------------------------------------------------------------
__global__ void fill_kernel(float* p, long n, float v) {
  long i = (long)blockIdx.x * blockDim.x + threadIdx.x;
  if (i < n) p[i] = v;
}

__global__ void deg_count_kernel(const int* __restrict__ dst, float* deg,
                                 long E) {
  long e = (long)blockIdx.x * blockDim.x + threadIdx.x;
  if (e < E) unsafeAtomicAdd(&deg[dst[e]], 1.0f);
}

__global__ void dinv_kernel(const float* __restrict__ deg, float* dinv,
                            long n) {
  long i = (long)blockIdx.x * blockDim.x + threadIdx.x;
  if (i < n) dinv[i] = rsqrtf(deg[i] + 1.0f);
}

// agg = xw * dinv[n]^2 + bias[c]   (self loop term + bias, init for scatter)
__global__ void agg_init_kernel(const float* __restrict__ xw,
                                const float* __restrict__ dinv,
                                const float* __restrict__ b, float* agg,
                                long n64) {
  long i = (long)blockIdx.x * blockDim.x + threadIdx.x;
  if (i >= n64) return;
  long node = i >> 6;
  int c = (int)(i & 63);
  float di = dinv[node];
  agg[i] = xw[i] * di * di + b[c];
}

// One wave per edge; each lane handles 2 channels (float2 gather + 2 fp32
// global atomics -> global_atomic_add_f32, L2-resident working set).
__global__ __launch_bounds__(256) void edge_scatter_kernel(
    const float* __restrict__ xw, const int* __restrict__ src,
    const int* __restrict__ dst, const float* __restrict__ dinv,
    float* __restrict__ agg, long E) {
  long t = (long)blockIdx.x * blockDim.x + threadIdx.x;
  long e = t >> 5;
  int lane = (int)(t & 31);
  if (e >= E) return;
  int s = src[e], d = dst[e];
  float coef = dinv[s] * dinv[d];
  const float2* xs = (const float2*)(xw + (long)s * 64);
  float2 v = xs[lane];
  float* ad = agg + (long)d * 64 + lane * 2;
  unsafeAtomicAdd(ad, v.x * coef);
  unsafeAtomicAdd(ad + 1, v.y * coef);
}

__global__ void mul_kernel(const float* a, const float* b, float* c, long n) {
  long i = (long)blockIdx.x * blockDim.x + threadIdx.x;
  if (i < n) c[i] = a[i] * b[i];
}

// H = Z*H + (1-Z)*Ht ; ACC += probs[t]*H
__global__ void gru_update_kernel(const float* __restrict__ Z,
                                  const float* __restrict__ HT, float* H,
                                  float* ACC, const float* __restrict__ probs,
                                  int t, long n) {
  long i = (long)blockIdx.x * blockDim.x + threadIdx.x;
  if (i >= n) return;
  float z = Z[i];
  float hn = z * H[i] + (1.0f - z) * HT[i];
  H[i] = hn;
  ACC[i] += probs[t] * hn;
}

__global__ void relu_kernel(const float* a, float* o, long n) {
  long i = (long)blockIdx.x * blockDim.x + threadIdx.x;
  if (i < n) o[i] = fmaxf(a[i], 0.0f);
}

__global__ void softmax12_kernel(const float* __restrict__ att, float* probs) {
  if (threadIdx.x == 0 && blockIdx.x == 0) {
    float m = att[0];
    for (int i = 1; i < GT; ++i) m = fmaxf(m, att[i]);
    float e[GT], s = 0.0f;
    for (int i = 0; i < GT; ++i) {
      e[i] = __expf(att[i] - m);
      s += e[i];
    }
    for (int i = 0; i < GT; ++i) probs[i] = e[i] / s;
  }
}

__global__ void label_count_kernel(const int* __restrict__ lab, float* cnt,
                                   long n) {
  long i = (long)blockIdx.x * blockDim.x + threadIdx.x;
  if (i < n) unsafeAtomicAdd(&cnt[lab[i]], 1.0f);
}

__global__ void group_sum64_kernel(const float* __restrict__ x,
                                   const int* __restrict__ lab, float* sum,
                                   long n64) {
  long i = (long)blockIdx.x * blockDim.x + threadIdx.x;
  if (i >= n64) return;
  long nidx = i >> 6;
  int c = (int)(i & 63);
  unsafeAtomicAdd(&sum[(long)lab[nidx] * 64 + c], x[i]);
}

// time[n][c] = x_1[n*96 + (C_IN-1)*T + c]
__global__ void group_sum_time_kernel(const float* __restrict__ x1,
                                      const int* __restrict__ lab, float* sum,
                                      long n12) {
  long i = (long)blockIdx.x * blockDim.x + threadIdx.x;
  if (i >= n12) return;
  long nidx = i / GT;
  int c = (int)(i - nidx * GT);
  unsafeAtomicAdd(&sum[(long)lab[nidx] * GT + c],
                  x1[nidx * (GC_IN * GT) + (GC_IN - 1) * GT + c]);
}

__global__ void combine_mean_kernel(const float* ss, const float* cs,
                                    const float* se, const float* ce,
                                    float* out, int oc, long n) {
  long i = (long)blockIdx.x * blockDim.x + threadIdx.x;
  if (i >= n) return;
  long m = i / oc;
  float a = ss[i] / fmaxf(cs[m], 1.0f);
  float b = se[i] / fmaxf(ce[m], 1.0f);
  out[i] = 0.5f * (a + b);
}

// x2cat[m, 0:16] = x_2[m]; x2cat[m, 16:80] = h_g[m]
__global__ void x2cat_kernel(const float* __restrict__ x2,
                             const float* __restrict__ hg, float* out,
                             long n80) {
  long i = (long)blockIdx.x * blockDim.x + threadIdx.x;
  if (i >= n80) return;
  long m = i / 80;
  int c = (int)(i - m * 80);
  out[i] = (c < GSECOND) ? x2[m * GSECOND + c] : hg[m * 64 + (c - GSECOND)];
}

// h_back[n] = 0.5*(h_[as[n]] + h_[ae[n]])
__global__ void hback_kernel(const float* __restrict__ hs,
                             const int* __restrict__ as_,
                             const int* __restrict__ ae_, float* out,
                             long n64) {
  long i = (long)blockIdx.x * blockDim.x + threadIdx.x;
  if (i >= n64) return;
  long nn = i >> 6;
  int c = (int)(i & 63);
  out[i] = 0.5f * (hs[(long)as_[nn] * 64 + c] + hs[(long)ae_[nn] * 64 + c]);
}

// ---------------------------------------------------------------------------
// Host-side orchestration
// ---------------------------------------------------------------------------
static inline unsigned gsz(long n, int b) { return (unsigned)((n + b - 1) / b); }

static void launch_gemm(hipStream_t st, const float* A1, int r1s, int c1s,
                        int K1, const float* A2, int r2s, int c2s, int K2,
                        const float* W, const float* bias, float* C, int N,
                        int OC, int act) {
  dim3 grid(N / 32), blk(32, 4);
  wmma_gemm_kernel<<<grid, blk, 0, st>>>(A1, r1s, c1s, K1, A2, r2s, c2s, K2, W,
                                         bias, C, OC, act);
}

struct XtDesc {  // per-time-step virtual A operand: base + t*tstep, strides
  const float* base;
  long tstep;
  int rs, cs, K;
};

static void run_a3tgcn(hipStream_t st, void* const* d_in, int pb, XtDesc A1d,
                       XtDesc A2d, int N, long E, const int* src,
                       const int* dst, const float* dinv, const float* probs,
                       float* Hb, float* ACCb, float* XWb, float* AGGb,
                       float* Zb, float* Rb, float* HRb, float* HTb,
                       float* out_relu) {
  const long n64 = (long)N * 64;
  fill_kernel<<<gsz(n64, 256), 256, 0, st>>>(Hb, n64, 0.0f);
  fill_kernel<<<gsz(n64, 256), 256, 0, st>>>(ACCb, n64, 0.0f);

  const float* Wg[3] = {(const float*)d_in[pb + 0], (const float*)d_in[pb + 4],
                        (const float*)d_in[pb + 8]};
  const float* bg[3] = {(const float*)d_in[pb + 1], (const float*)d_in[pb + 5],
                        (const float*)d_in[pb + 9]};
  const float* Wl[3] = {(const float*)d_in[pb + 2], (const float*)d_in[pb + 6],
                        (const float*)d_in[pb + 10]};
  const float* bl[3] = {(const float*)d_in[pb + 3], (const float*)d_in[pb + 7],
                        (const float*)d_in[pb + 11]};

  for (int t = 0; t < GT; ++t) {
    const float* xt1 = A1d.base + A1d.tstep * t;
    const float* xt2 = A2d.base ? (A2d.base + A2d.tstep * t) : nullptr;
    for (int g = 0; g < 3; ++g) {
      // xw = xt @ Wg
      launch_gemm(st, xt1, A1d.rs, A1d.cs, A1d.K, xt2, A2d.rs, A2d.cs, A2d.K,
                  Wg[g], nullptr, XWb, N, 64, ACT_NONE);
      // agg = xw*dinv^2 + bg, then scatter edges
      agg_init_kernel<<<gsz(n64, 256), 256, 0, st>>>(XWb, dinv, bg[g], AGGb,
                                                     n64);
      edge_scatter_kernel<<<gsz(E * 32, 256), 256, 0, st>>>(XWb, src, dst,
                                                            dinv, AGGb, E);
      if (g == 0) {  // Z = sigmoid([agg, H] @ Wl_z + bl_z)
        launch_gemm(st, AGGb, 64, 1, 64, Hb, 64, 1, 64, Wl[0], bl[0], Zb, N,
                    64, ACT_SIGMOID);
      } else if (g == 1) {  // R, then HR = H*R
        launch_gemm(st, AGGb, 64, 1, 64, Hb, 64, 1, 64, Wl[1], bl[1], Rb, N,
                    64, ACT_SIGMOID);
        mul_kernel<<<gsz(n64, 256), 256, 0, st>>>(Hb, Rb, HRb, n64);
      } else {  // Ht = tanh([agg, H*R] @ Wl_h + bl_h)
        launch_gemm(st, AGGb, 64, 1, 64, HRb, 64, 1, 64, Wl[2], bl[2], HTb, N,
                    64, ACT_TANH);
      }
    }
    gru_update_kernel<<<gsz(n64, 256), 256, 0, st>>>(Zb, HTb, Hb, ACCb, probs,
                                                     t, n64);
  }
  relu_kernel<<<gsz(n64, 256), 256, 0, st>>>(ACCb, out_relu, n64);
}

// Input index map (setup_inputs() dict insertion order, recursed into params):
//  main_1: 0..12  [Wg_z,bg_z,Wl_z,bl_z, Wg_r,bg_r,Wl_r,bl_r, Wg_h,bg_h,Wl_h,bl_h, att]
//  main_2: 13..25   main_3: 26..38
//  39 lin1_W  40 lin1_b  41 lin2_W  42 lin2_b
//  43 x_1  44 edge_index_1  45 x_2  46 edge_index_2  47 addr_start  48 addr_end
extern "C" void kernel_launch(void* const* d_in, const int* in_sizes, int n_in,
                              void* d_out, int out_size, void* d_ws,
                              size_t ws_size, hipStream_t stream) {
  const float* x1 = (const float*)d_in[43];
  const int* ei1 = (const int*)d_in[44];
  const float* x2 = (const float*)d_in[45];
  const int* ei2 = (const int*)d_in[46];
  const int* as_ = (const int*)d_in[47];
  const int* ae_ = (const int*)d_in[48];
  const long E1 = in_sizes[44] / 2, E2 = in_sizes[46] / 2;
  const int* src1 = ei1;
  const int* dst1 = ei1 + E1;
  const int* src2 = ei2;
  const int* dst2 = ei2 + E2;

  // --- workspace carve ---
  char* w = (char*)d_ws;
  auto alloc = [&](size_t bytes) {
    char* p = w;
    w += (bytes + 255) & ~(size_t)255;
    return (float*)p;
  };
  const size_t big = (size_t)GN1 * 64 * sizeof(float);
  float* Hb = alloc(big);
  float* ACCb = alloc(big);
  float* XWb = alloc(big);
  float* AGGb = alloc(big);
  float* Zb = alloc(big);
  float* Rb = alloc(big);
  float* HRb = alloc(big);
  float* HTb = alloc(big);
  float* h_buf = alloc(big);   // relu(a3tgcn main_1)
  float* h2_buf = alloc(big);  // relu(a3tgcn main_2)
  float* hback = alloc(big);
  float* hh_buf = alloc(big);
  float* dinv1 = alloc((size_t)GN1 * 4);
  float* deg = alloc((size_t)GN1 * 4);
  float* dinv2 = alloc((size_t)GN2 * 4);
  float* probs1 = alloc(GT * 4);
  float* probs2 = alloc(GT * 4);
  float* probs3 = alloc(GT * 4);
  float* cnt_s = alloc((size_t)GN2 * 4);
  float* cnt_e = alloc((size_t)GN2 * 4);
  float* sum_s = alloc((size_t)GN2 * 64 * 4);
  float* sum_e = alloc((size_t)GN2 * 64 * 4);
  float* hg = alloc((size_t)GN2 * 64 * 4);
  float* sum_ts = alloc((size_t)GN2 * GT * 4);
  float* sum_te = alloc((size_t)GN2 * GT * 4);
  float* h_time = alloc((size_t)GN2 * GT * 4);
  float* x2cat = alloc((size_t)GN2 * 80 * 4);
  float* hsmall = alloc((size_t)GN2 * 64 * 4);  // relu(a3tgcn main_3)

  // --- attention softmax (3x) ---
  softmax12_kernel<<<1, 32, 0, stream>>>((const float*)d_in[12], probs1);
  softmax12_kernel<<<1, 32, 0, stream>>>((const float*)d_in[25], probs2);
  softmax12_kernel<<<1, 32, 0, stream>>>((const float*)d_in[38], probs3);

  // --- deg^{-1/2} for both graphs ---
  fill_kernel<<<gsz(GN1, 256), 256, 0, stream>>>(deg, GN1, 0.0f);
  deg_count_kernel<<<gsz(E1, 256), 256, 0, stream>>>(dst1, deg, E1);
  dinv_kernel<<<gsz(GN1, 256), 256, 0, stream>>>(deg, dinv1, GN1);
  fill_kernel<<<gsz(GN2, 256), 256, 0, stream>>>(deg, GN2, 0.0f);
  deg_count_kernel<<<gsz(E2, 256), 256, 0, stream>>>(dst2, deg, E2);
  dinv_kernel<<<gsz(GN2, 256), 256, 0, stream>>>(deg, dinv2, GN2);

  // --- a3tgcn main_1: xt = x_1[:, :, t]  (row stride 96, col stride 12) ---
  XtDesc a1 = {x1, 1, GC_IN * GT, GT, GC_IN};
  XtDesc none = {nullptr, 0, 0, 0, 0};
  run_a3tgcn(stream, d_in, 0, a1, none, GN1, E1, src1, dst1, dinv1, probs1, Hb,
             ACCb, XWb, AGGb, Zb, Rb, HRb, HTb, h_buf);

  // --- group means: h_g and h_time ---
  fill_kernel<<<gsz(GN2, 256), 256, 0, stream>>>(cnt_s, GN2, 0.0f);
  fill_kernel<<<gsz(GN2, 256), 256, 0, stream>>>(cnt_e, GN2, 0.0f);
  label_count_kernel<<<gsz(GN1, 256), 256, 0, stream>>>(as_, cnt_s, GN1);
  label_count_kernel<<<gsz(GN1, 256), 256, 0, stream>>>(ae_, cnt_e, GN1);
  const long n2_64 = (long)GN2 * 64, n2_12 = (long)GN2 * GT;
  fill_kernel<<<gsz(n2_64, 256), 256, 0, stream>>>(sum_s, n2_64, 0.0f);
  fill_kernel<<<gsz(n2_64, 256), 256, 0, stream>>>(sum_e, n2_64, 0.0f);
  group_sum64_kernel<<<gsz((long)GN1 * 64, 256), 256, 0, stream>>>(
      h_buf, as_, sum_s, (long)GN1 * 64);
  group_sum64_kernel<<<gsz((long)GN1 * 64, 256), 256, 0, stream>>>(
      h_buf, ae_, sum_e, (long)GN1 * 64);
  combine_mean_kernel<<<gsz(n2_64, 256), 256, 0, stream>>>(
      sum_s, cnt_s, sum_e, cnt_e, hg, 64, n2_64);
  fill_kernel<<<gsz(n2_12, 256), 256, 0, stream>>>(sum_ts, n2_12, 0.0f);
  fill_kernel<<<gsz(n2_12, 256), 256, 0, stream>>>(sum_te, n2_12, 0.0f);
  group_sum_time_kernel<<<gsz((long)GN1 * GT, 256), 256, 0, stream>>>(
      x1, as_, sum_ts, (long)GN1 * GT);
  group_sum_time_kernel<<<gsz((long)GN1 * GT, 256), 256, 0, stream>>>(
      x1, ae_, sum_te, (long)GN1 * GT);
  combine_mean_kernel<<<gsz(n2_12, 256), 256, 0, stream>>>(
      sum_ts, cnt_s, sum_te, cnt_e, h_time, GT, n2_12);

  // --- x2cat = [x_2 | h_g]  (N2 x 80), then a3tgcn main_3 on graph 2 ---
  x2cat_kernel<<<gsz((long)GN2 * 80, 256), 256, 0, stream>>>(
      x2, hg, x2cat, (long)GN2 * 80);
  XtDesc a3_1 = {x2cat, 0, 80, 1, 80};
  XtDesc a3_2 = {h_time, 1, GT, 1, 1};  // h_time[:, t] column
  run_a3tgcn(stream, d_in, 26, a3_1, a3_2, GN2, E2, src2, dst2, dinv2, probs3,
             Hb, ACCb, XWb, AGGb, Zb, Rb, HRb, HTb, hsmall);

  // --- a3tgcn main_2 on graph 1: xt = [h | time[:, t]]  (K = 65) ---
  XtDesc a2_1 = {h_buf, 0, 64, 1, 64};
  XtDesc a2_2 = {x1 + (GC_IN - 1) * GT, 1, GC_IN * GT, 1, 1};  // time column
  run_a3tgcn(stream, d_in, 13, a2_1, a2_2, GN1, E1, src1, dst1, dinv1, probs2,
             Hb, ACCb, XWb, AGGb, Zb, Rb, HRb, HTb, h2_buf);

  // --- h_back = 0.5*(h_[as] + h_[ae]) ---
  hback_kernel<<<gsz((long)GN1 * 64, 256), 256, 0, stream>>>(
      hsmall, as_, ae_, hback, (long)GN1 * 64);

  // --- hh = relu([h2 | h_back] @ lin1_W + lin1_b);  out = hh @ lin2_W + b ---
  launch_gemm(stream, h2_buf, 64, 1, 64, hback, 64, 1, 64,
              (const float*)d_in[39], (const float*)d_in[40], hh_buf, GN1, 64,
              ACT_RELU);
  launch_gemm(stream, hh_buf, 64, 1, 64, nullptr, 0, 0, 0,
              (const float*)d_in[41], (const float*)d_in[42], (float*)d_out,
              GN1, GT, ACT_NONE);
}